// graph_conv_87668872446777
// MI455X (gfx1250) — compile-verified
//
#include <hip/hip_runtime.h>

typedef __attribute__((ext_vector_type(2))) float v2f;
typedef __attribute__((ext_vector_type(8))) float v8f;

#define BB 8
#define NN 4096
#define NFACES 8192
#define KNB 16
#define FIN 64
#define TT 17
#define KKTOT 1088        // TT*FIN
#define FOUT 128
#define FOUT2 256
#define HKK 544           // half of KKTOT, staged per phase
#define LDSTR 548         // padded LDS row stride (548%64=36 -> conflict free)
#define SBLK 64           // stats reduction blocks

// ---------------- adjacency build: 6 atomicOr bits per face ----------------
__global__ __launch_bounds__(256) void buildAdj(const long long* faces, unsigned* adjBits) {
    int i = blockIdx.x * blockDim.x + threadIdx.x;   // b*NFACES + f
    if (i >= BB * NFACES) return;
    int b = i / NFACES;
    const long long* fp = faces + (size_t)i * 3;
    int a0 = (int)fp[0], a1 = (int)fp[1], a2 = (int)fp[2];
    int mn = min(a0, min(a1, a2));
    int mx = max(a0, max(a1, a2));
    int md = a0 + a1 + a2 - mn - mx;
    unsigned* base = adjBits + (size_t)b * NN * (NN / 32);
    #define SETB(r, c) atomicOr(&base[(size_t)(r) * (NN/32) + ((c) >> 5)], 1u << ((c) & 31))
    SETB(mn, md); SETB(md, mn);
    SETB(mn, mx); SETB(mx, mn);
    SETB(md, mx); SETB(mx, md);
    #undef SETB
}

// ---- extract 16 smallest-index neighbors per node (== top_k tie semantics) ----
__global__ __launch_bounds__(128) void extractNbrs(const unsigned* adjBits, int* nbrIdx) {
    int i = blockIdx.x * blockDim.x + threadIdx.x;   // b*N + n
    if (i >= BB * NN) return;
    const unsigned* row = adjBits + (size_t)i * (NN / 32);
    int* out = nbrIdx + (size_t)i * KNB;
    int cnt = 0;
    for (int w = 0; w < NN / 32 && cnt < KNB; ++w) {
        unsigned bits = row[w];
        while (bits && cnt < KNB) {
            int t = __builtin_ctz(bits);
            out[cnt++] = w * 32 + t;
            bits &= bits - 1;
        }
    }
    for (; cnt < KNB; ++cnt) out[cnt] = -1;
}

// ---- W transpose, K-pair interleaved: Wt2[kk/2][o][2], kk = t*64 + f ----
// element (kk, o) lives at (kk>>1)*2*FOUT + o*2 + (kk&1)
__global__ __launch_bounds__(256) void makeWt(const float* conv_w, float* Wt2) {
    int i = blockIdx.x * blockDim.x + threadIdx.x;   // kk*FOUT + o
    if (i >= KKTOT * FOUT) return;
    int kk = i / FOUT, o = i % FOUT;
    int t = kk >> 6, f = kk & 63;
    float v = conv_w[(size_t)o * (FIN * TT) + (size_t)f * TT + t];
    Wt2[(size_t)(kk >> 1) * (2 * FOUT) + o * 2 + (kk & 1)] = v;
}

// ---------------- main GEMM: h[16 nodes][128] via V_WMMA_F32_16X16X4_F32 ----------------
__global__ __launch_bounds__(128) void conv1(const float* __restrict__ x,
                                             const int* __restrict__ nbrIdx,
                                             const float* __restrict__ Wt2,
                                             const float* __restrict__ conv_b,
                                             float* __restrict__ h) {
    __shared__ float feat[16 * LDSTR];
    int wg = blockIdx.x;
    int bb = wg / (NN / 16);
    int nbase = (wg % (NN / 16)) * 16;
    int tid = threadIdx.x;
    int wave = tid >> 5, lane = tid & 31;
    int mrow = lane & 15;
    int khalf = (lane >> 4) << 1;       // 0 or 2
    int mhalf = (lane >> 4) << 3;       // 0 or 8 (D-matrix row offset)
    int ct0 = wave * 2, ct1 = wave * 2 + 1;
    int col0 = ct0 * 16 + mrow;
    int col1 = ct1 * 16 + mrow;

    v8f acc0 = {}; v8f acc1 = {};
    const float* xb = x + (size_t)bb * NN * FIN;

    for (int phase = 0; phase < 2; ++phase) {
        int kk0 = phase * HKK;
        __syncthreads();
        // cooperative gather of the 16 x 544 feat slice into LDS
        for (int e = tid; e < 16 * HKK; e += 128) {
            int m = e / HKK;
            int kk = e - m * HKK;
            int kkg = kk0 + kk;
            int t = kkg >> 6, f = kkg & 63;
            float v;
            if (t == 0) {
                v = xb[(size_t)(nbase + m) * FIN + f];
            } else {
                int idx = nbrIdx[((size_t)bb * NN + nbase + m) * KNB + (t - 1)];
                v = (idx >= 0) ? xb[(size_t)idx * FIN + f] : 0.0f;
            }
            feat[m * LDSTR + kk] = v;
        }
        __syncthreads();
        for (int kb = 0; kb < HKK; kb += 4) {
            int kg = kk0 + kb + khalf;        // even
            // A fragment: lane holds feat[mrow][kb+khalf .. +1]  (8B aligned LDS load)
            v2f a = *(const v2f*)&feat[mrow * LDSTR + kb + khalf];
            // B fragments: one coalesced 8B load each (rows kg, kg+1 interleaved)
            const float* wrow = Wt2 + (size_t)(kg >> 1) * (2 * FOUT);
            v2f b0 = *(const v2f*)&wrow[col0 * 2];
            v2f b1 = *(const v2f*)&wrow[col1 * 2];
            acc0 = __builtin_amdgcn_wmma_f32_16x16x4_f32(false, a, false, b0,
                                                         (short)0, acc0, false, false);
            acc1 = __builtin_amdgcn_wmma_f32_16x16x4_f32(false, a, false, b1,
                                                         (short)0, acc1, false, false);
        }
    }
    // D layout: vgpr i, lane -> row (i + mhalf), col mrow
    float bias0 = conv_b[col0];
    float bias1 = conv_b[col1];
    #pragma unroll
    for (int i = 0; i < 8; ++i) {
        size_t node = (size_t)bb * NN + nbase + i + mhalf;
        h[node * FOUT + col0] = acc0[i] + bias0;
        h[node * FOUT + col1] = acc1[i] + bias1;
    }
}

// ---------------- deterministic per-channel partial stats over h ----------------
__global__ __launch_bounds__(128) void stats1k(const float* __restrict__ h, float* __restrict__ stats) {
    int o = threadIdx.x;                      // channel
    size_t r0 = (size_t)blockIdx.x * ((size_t)BB * NN / SBLK);
    float s = 0.f, s2 = 0.f;
    for (size_t r = r0; r < r0 + (size_t)BB * NN / SBLK; ++r) {
        float v = h[r * FOUT + o];
        s += v; s2 += v * v;
    }
    stats[((size_t)blockIdx.x * 2) * FOUT + o]     = s;
    stats[((size_t)blockIdx.x * 2 + 1) * FOUT + o] = s2;
}

__global__ void finalize1(const float* stats, const float* gamma, const float* beta, float* mv) {
    int o = threadIdx.x;
    if (o >= FOUT) return;
    float s = 0.f, s2 = 0.f;
    for (int blk = 0; blk < SBLK; ++blk) {
        s  += stats[((size_t)blk * 2) * FOUT + o];
        s2 += stats[((size_t)blk * 2 + 1) * FOUT + o];
    }
    float inv = 1.0f / (float)(BB * NN);
    float mu = s * inv;
    float var = s2 * inv - mu * mu;
    float sc = rsqrtf(var + 1e-5f) * gamma[o];
    mv[o] = sc;
    mv[FOUT + o] = beta[o] - mu * sc;
}

// ---------------- BN1 + ReLU + maxpool(16) + conv2 (8->256) ----------------
__global__ __launch_bounds__(256) void poolConv2(const float* __restrict__ h,
                                                 const float* __restrict__ mv1,
                                                 const float* __restrict__ w2,
                                                 const float* __restrict__ b2,
                                                 float* __restrict__ y) {
    __shared__ float hb[FOUT];
    __shared__ float hp[FOUT / KNB];
    size_t node = blockIdx.x;
    int tid = threadIdx.x;
    if (tid < FOUT) {
        float v = h[node * FOUT + tid] * mv1[tid] + mv1[FOUT + tid];
        hb[tid] = fmaxf(v, 0.0f);
    }
    __syncthreads();
    if (tid < FOUT / KNB) {
        float m = hb[tid * KNB];
        #pragma unroll
        for (int j = 1; j < KNB; ++j) m = fmaxf(m, hb[tid * KNB + j]);
        hp[tid] = m;
    }
    __syncthreads();
    float acc = b2[tid];
    #pragma unroll
    for (int c = 0; c < FOUT / KNB; ++c) acc += hp[c] * w2[tid * (FOUT / KNB) + c];
    y[node * FOUT2 + tid] = acc;
}

__global__ __launch_bounds__(256) void stats2k(const float* __restrict__ y, float* __restrict__ stats) {
    int o = threadIdx.x;
    size_t r0 = (size_t)blockIdx.x * ((size_t)BB * NN / SBLK);
    float s = 0.f, s2 = 0.f;
    for (size_t r = r0; r < r0 + (size_t)BB * NN / SBLK; ++r) {
        float v = y[r * FOUT2 + o];
        s += v; s2 += v * v;
    }
    stats[((size_t)blockIdx.x * 2) * FOUT2 + o]     = s;
    stats[((size_t)blockIdx.x * 2 + 1) * FOUT2 + o] = s2;
}

__global__ void finalize2(const float* stats, const float* gamma, const float* beta, float* mv) {
    int o = threadIdx.x;
    if (o >= FOUT2) return;
    float s = 0.f, s2 = 0.f;
    for (int blk = 0; blk < SBLK; ++blk) {
        s  += stats[((size_t)blk * 2) * FOUT2 + o];
        s2 += stats[((size_t)blk * 2 + 1) * FOUT2 + o];
    }
    float inv = 1.0f / (float)(BB * NN);
    float mu = s * inv;
    float var = s2 * inv - mu * mu;
    float sc = rsqrtf(var + 1e-5f) * gamma[o];
    mv[o] = sc;
    mv[FOUT2 + o] = beta[o] - mu * sc;
}

__global__ __launch_bounds__(256) void bn2relu(float* __restrict__ y, const float* __restrict__ mv) {
    size_t i = (size_t)blockIdx.x * blockDim.x + threadIdx.x;
    if (i >= (size_t)BB * NN * FOUT2) return;
    int o = (int)(i % FOUT2);
    y[i] = fmaxf(y[i] * mv[o] + mv[FOUT2 + o], 0.0f);
}

extern "C" void kernel_launch(void* const* d_in, const int* in_sizes, int n_in,
                              void* d_out, int out_size, void* d_ws, size_t ws_size,
                              hipStream_t stream) {
    const float*     x      = (const float*)d_in[0];
    const long long* faces  = (const long long*)d_in[1];   // int64
    const float*     conv_w = (const float*)d_in[2];
    const float*     conv_b = (const float*)d_in[3];
    const float*     bn_g   = (const float*)d_in[4];
    const float*     bn_b   = (const float*)d_in[5];
    const float*     w2     = (const float*)d_in[6];
    const float*     b2     = (const float*)d_in[7];
    const float*     bn2_g  = (const float*)d_in[8];
    const float*     bn2_b  = (const float*)d_in[9];
    float* y = (float*)d_out;

    char* ws = (char*)d_ws;
    unsigned* adjBits = (unsigned*)(ws);                          // 16 MB
    int*      nbrIdx  = (int*)     (ws + ((size_t)16 << 20));     // 2 MB
    float*    Wt2     = (float*)   (ws + ((size_t)18 << 20));     // 0.56 MB
    float*    h       = (float*)   (ws + ((size_t)19 << 20));     // 16 MB
    float*    stats1  = (float*)   (ws + ((size_t)35 << 20));                 // 64 KB
    float*    stats2  = (float*)   (ws + ((size_t)35 << 20) + (64u << 10));   // 128 KB
    float*    mv1     = (float*)   (ws + ((size_t)35 << 20) + (192u << 10));  // 1 KB
    float*    mv2     = (float*)   (ws + ((size_t)35 << 20) + (196u << 10));  // 2 KB

    hipMemsetAsync(adjBits, 0, (size_t)16 << 20, stream);

    buildAdj   <<<(BB * NFACES + 255) / 256, 256, 0, stream>>>(faces, adjBits);
    extractNbrs<<<(BB * NN + 127) / 128,    128, 0, stream>>>(adjBits, nbrIdx);
    makeWt     <<<(KKTOT * FOUT + 255)/256, 256, 0, stream>>>(conv_w, Wt2);
    conv1      <<<BB * NN / 16,             128, 0, stream>>>(x, nbrIdx, Wt2, conv_b, h);
    stats1k    <<<SBLK,                     128, 0, stream>>>(h, stats1);
    finalize1  <<<1,                        128, 0, stream>>>(stats1, bn_g, bn_b, mv1);
    poolConv2  <<<BB * NN,                  256, 0, stream>>>(h, mv1, w2, b2, y);
    stats2k    <<<SBLK,                     256, 0, stream>>>(y, stats2);
    finalize2  <<<1,                        256, 0, stream>>>(stats2, bn2_g, bn2_b, mv2);
    bn2relu    <<<(BB * NN * FOUT2 + 255)/256, 256, 0, stream>>>(y, mv2);
}